// RoIAlign_46127948759718
// MI455X (gfx1250) — compile-verified
//
#include <hip/hip_runtime.h>
#include <stdint.h>

// Feature map geometry (fixed by the reference harness).
static constexpr int FW = 256;      // feature width
static constexpr int FH = 256;      // feature height
static constexpr int FC = 256;      // channels
static constexpr int PHB = 7;       // pooled height
static constexpr int PWB = 7;       // pooled width
static constexpr int CCHUNK = 32;   // channels per block (grid.y = FC/CCHUNK)
// Per-buffer LDS tile floats; 2 buffers x 32000 B = 64000 B (< 64 KB static LDS).
// Max ROI tile is ~62x62 = 3844 floats, so nc >= 2 always on the TDM path.
static constexpr int TILE_FLOATS = 8000;
#define ROI_SCALE (1.0f / 32.0f)

typedef unsigned int v4u __attribute__((ext_vector_type(4)));
typedef int          v8i __attribute__((ext_vector_type(8)));
typedef int          v4i __attribute__((ext_vector_type(4)));

// Issue one TDM tensor load: 3D tile (tile_w x tile_h x nc) of 4-byte elements
// from global (row stride FW, plane stride FH*FW) into LDS at lds_byte,
// packed contiguously as [c][y][x].
// D# packing per CDNA5 ISA.
// This toolchain (clang-23 / therock-10.0) exposes the 6-arg builtin:
//   (u32x4 group0, i32x8 group1, i32x4 group2, i32x4 group3, i32x8 extra, i32 cpol)
static __device__ __forceinline__ void tdm_tile_load(unsigned glo, unsigned ghi,
                                                     unsigned lds_byte,
                                                     int tw, int th, int nc) {
  unsigned tw16 = (unsigned)tw & 0xFFFFu;
  unsigned th16 = (unsigned)th & 0xFFFFu;
  v4u g0 = {
      1u,                                   // count=1 (valid), user mode, no gather
      lds_byte,                             // lds_addr
      glo,                                  // global_addr[31:0]
      (ghi & 0x01FFFFFFu) | (2u << 30)      // global_addr[56:32] | type=2 (image)
  };
  v8i g1 = {
      (int)(2u << 16),                                    // wg_mask=0, data_size=2 (4B), no pad/iter
      (int)(tw16 << 16),                                  // atomic_addr=0 | tensor_dim0[15:0]
      (int)((((unsigned)tw >> 16) & 0xFFFFu) | (th16 << 16)), // tensor_dim0[31:16] | tensor_dim1[15:0]
      (int)((((unsigned)th >> 16) & 0xFFFFu) | (tw16 << 16)), // tensor_dim1[31:16] | tile_dim0
      (int)(th16 | (((unsigned)nc & 0xFFFFu) << 16)),     // tile_dim1 | tile_dim2
      FW,                                                 // tensor_dim0_stride[31:0] = 256
      0,                                                  // stride0[47:32]=0 | stride1[15:0]=0
      1                                                   // stride1[47:16]: 65536>>16 = 1
  };
  v4i g2 = { nc, 0, 0, 0 };   // tensor_dim2=nc, dim3 unused, dim2_stride unused, tile_dim3=0
  v4i g3 = { 0, 0, 0, 0 };
  v8i gx = { 0, 0, 0, 0, 0, 0, 0, 0 };  // uncharacterized operand on this toolchain
  __builtin_amdgcn_tensor_load_to_lds(g0, g1, g2, g3, gx, 0);
}

__global__ __launch_bounds__(256) void roi_align_tdm_kernel(
    const float* __restrict__ feat, const float* __restrict__ rois,
    float* __restrict__ out) {
  __shared__ int   s_x0[14], s_y0[14];
  __shared__ float s_wx[14], s_wy[14];
  __shared__ int   s_vx[PWB], s_vy[PHB];
  __shared__ float s_tile[2 * TILE_FLOATS];

  const int r  = blockIdx.x;
  const int c0 = blockIdx.y * CCHUNK;
  const int t  = threadIdx.x;

  // --- Per-ROI scalars (recomputed by every thread; VALU-cheap) ---
  const float* roi = rois + (size_t)r * 5;
  const int   b  = (int)roi[0];
  const float x1 = fminf(fmaxf(roi[1] * ROI_SCALE, 0.0f), (float)(FW - 1));
  const float y1 = fminf(fmaxf(roi[2] * ROI_SCALE, 0.0f), (float)(FH - 1));
  const float x2 = fminf(fmaxf(roi[3] * ROI_SCALE, 0.0f), (float)(FW - 1));
  const float y2 = fminf(fmaxf(roi[4] * ROI_SCALE, 0.0f), (float)(FH - 1));
  const float bin_w = (x2 - x1) / (float)PWB;
  const float bin_h = (y2 - y1) / (float)PHB;
  const float sub_w = bin_w * 0.5f;
  const float sub_h = bin_h * 0.5f;

  // --- Sample grid: 14 x-coords, 14 y-coords, per-bin validity ---
  if (t < 14) {
    int pw = t >> 1, kx = t & 1;
    float bsx = fminf(fmaxf(x1 + (float)pw * bin_w, 0.0f), (float)(FW - 1));
    float X = bsx + sub_w * 0.5f + (float)kx * sub_w;
    int x0 = (int)floorf(X);
    x0 = x0 < 0 ? 0 : (x0 > FW - 2 ? FW - 2 : x0);
    s_x0[t] = x0;
    s_wx[t] = X - (float)x0;
  } else if (t < 28) {
    int u = t - 14;
    int ph = u >> 1, ky = u & 1;
    float bsy = fminf(fmaxf(y1 + (float)ph * bin_h, 0.0f), (float)(FH - 1));
    float Y = bsy + sub_h * 0.5f + (float)ky * sub_h;
    int y0 = (int)floorf(Y);
    y0 = y0 < 0 ? 0 : (y0 > FH - 2 ? FH - 2 : y0);
    s_y0[u] = y0;
    s_wy[u] = Y - (float)y0;
  } else if (t < 35) {
    int pw = t - 28;
    float bsx = fminf(fmaxf(x1 + (float)pw * bin_w, 0.0f), (float)(FW - 1));
    float bex = fminf(fmaxf(x1 + (float)(pw + 1) * bin_w, 0.0f), (float)(FW - 1));
    s_vx[pw] = bsx < bex;
  } else if (t < 42) {
    int ph = t - 35;
    float bsy = fminf(fmaxf(y1 + (float)ph * bin_h, 0.0f), (float)(FH - 1));
    float bey = fminf(fmaxf(y1 + (float)(ph + 1) * bin_h, 0.0f), (float)(FH - 1));
    s_vy[ph] = bsy < bey;
  }
  __syncthreads();

  // Tile bounds: sample coords are min at index 0, max at index 13 (monotone).
  const int tx0 = s_x0[0], ty0 = s_y0[0];
  const int tw  = s_x0[13] + 2 - tx0;
  const int th  = s_y0[13] + 2 - ty0;
  const int tsz = tw * th;
  float* outr = out + ((size_t)r * FC + c0) * (PHB * PWB);

  if (tsz <= TILE_FLOATS) {
    // ---------- TDM-staged path ----------
    int nc = TILE_FLOATS / tsz;
    if (nc > CCHUNK) nc = CCHUNK;
    const int nsteps = (CCHUNK + nc - 1) / nc;

    const uint64_t gbase = (uint64_t)(uintptr_t)feat +
        4ull * ((uint64_t)(b * FC + c0) * (uint64_t)(FH * FW) +
                (uint64_t)(ty0 * FW + tx0));

    if (t < warpSize) {  // wave 0 owns TDM issue; TENSORcnt is per-wave
      unsigned glo = (unsigned)__builtin_amdgcn_readfirstlane((int)(unsigned)gbase);
      unsigned ghi = (unsigned)__builtin_amdgcn_readfirstlane((int)(unsigned)(gbase >> 32));
      unsigned l0  = (unsigned)__builtin_amdgcn_readfirstlane(
                         (int)(unsigned)(uintptr_t)(&s_tile[0]));
      int stw = __builtin_amdgcn_readfirstlane(tw);
      int sth = __builtin_amdgcn_readfirstlane(th);
      int sn0 = __builtin_amdgcn_readfirstlane(nc);
      tdm_tile_load(glo, ghi, l0, stw, sth, sn0);
    }

    for (int s = 0; s < nsteps; ++s) {
      const float* buf = &s_tile[(s & 1) * TILE_FLOATS];
      const int cs  = s * nc;
      int cnt = CCHUNK - cs;
      if (cnt > nc) cnt = nc;

      if (t < warpSize) {
        if (s + 1 < nsteps) {
          // Double-buffer: issue next before waiting for current (in-order per wave).
          int cs1  = (s + 1) * nc;
          int cnt1 = CCHUNK - cs1;
          if (cnt1 > nc) cnt1 = nc;
          uint64_t ga = gbase + 4ull * (uint64_t)cs1 * (uint64_t)(FH * FW);
          unsigned glo = (unsigned)__builtin_amdgcn_readfirstlane((int)(unsigned)ga);
          unsigned ghi = (unsigned)__builtin_amdgcn_readfirstlane((int)(unsigned)(ga >> 32));
          unsigned l1  = (unsigned)__builtin_amdgcn_readfirstlane(
                             (int)(unsigned)(uintptr_t)(&s_tile[((s + 1) & 1) * TILE_FLOATS]));
          int stw = __builtin_amdgcn_readfirstlane(tw);
          int sth = __builtin_amdgcn_readfirstlane(th);
          int sn1 = __builtin_amdgcn_readfirstlane(cnt1);
          tdm_tile_load(glo, ghi, l1, stw, sth, sn1);
          __builtin_amdgcn_s_wait_tensorcnt(1);  // current tile done
        } else {
          __builtin_amdgcn_s_wait_tensorcnt(0);
        }
      }
      __syncthreads();  // tile for step s visible to all waves

      for (int idx = t; idx < cnt * PHB * PWB; idx += (int)blockDim.x) {
        int cl  = idx / 49;
        int bin = idx - cl * 49;
        int ph  = bin / 7;
        int pw  = bin - ph * 7;
        float res = 0.0f;
        if (s_vx[pw] & s_vy[ph]) {
          const float* cbuf = buf + cl * tsz;
          float m = -3.402823466e+38f;
#pragma unroll
          for (int ky = 0; ky < 2; ++ky) {
            int uy = ph * 2 + ky;
            int yl = s_y0[uy] - ty0;
            float wy = s_wy[uy];
#pragma unroll
            for (int kx = 0; kx < 2; ++kx) {
              int ux = pw * 2 + kx;
              int xl = s_x0[ux] - tx0;
              float wx = s_wx[ux];
              const float* p = cbuf + yl * tw + xl;
              float lt = p[0], rt = p[1], lb = p[tw], rb = p[tw + 1];
              float mt = wx * rt + (1.0f - wx) * lt;
              float mb = wx * rb + (1.0f - wx) * lb;
              float pix = wy * mb + (1.0f - wy) * mt;
              m = fmaxf(m, pix);
            }
          }
          res = m;
        }
        outr[(size_t)(cs + cl) * 49 + bin] = res;
      }
      __syncthreads();  // reads done before this buffer is overwritten at s+2
    }
  } else {
    // ---------- Fallback: direct global gather (oversized ROI tile) ----------
    const float* fb = feat + (size_t)(b * FC + c0) * (FH * FW);
    for (int idx = t; idx < CCHUNK * PHB * PWB; idx += (int)blockDim.x) {
      int cl  = idx / 49;
      int bin = idx - cl * 49;
      int ph  = bin / 7;
      int pw  = bin - ph * 7;
      float res = 0.0f;
      if (s_vx[pw] & s_vy[ph]) {
        const float* cp = fb + (size_t)cl * (FH * FW);
        float m = -3.402823466e+38f;
#pragma unroll
        for (int ky = 0; ky < 2; ++ky) {
          int uy = ph * 2 + ky;
          int y0 = s_y0[uy];
          float wy = s_wy[uy];
#pragma unroll
          for (int kx = 0; kx < 2; ++kx) {
            int ux = pw * 2 + kx;
            int x0 = s_x0[ux];
            float wx = s_wx[ux];
            const float* p = cp + y0 * FW + x0;
            float lt = p[0], rt = p[1], lb = p[FW], rb = p[FW + 1];
            float mt = wx * rt + (1.0f - wx) * lt;
            float mb = wx * rb + (1.0f - wx) * lb;
            m = fmaxf(m, wy * mb + (1.0f - wy) * mt);
          }
        }
        res = m;
      }
      outr[(size_t)cl * 49 + bin] = res;
    }
  }
}

extern "C" void kernel_launch(void* const* d_in, const int* in_sizes, int n_in,
                              void* d_out, int out_size, void* d_ws, size_t ws_size,
                              hipStream_t stream) {
  (void)n_in; (void)out_size; (void)d_ws; (void)ws_size;
  const float* feat = (const float*)d_in[0];
  const float* rois = (const float*)d_in[1];
  float* out = (float*)d_out;
  const int num_roi = in_sizes[1] / 5;
  dim3 grid((unsigned)num_roi, FC / CCHUNK, 1);
  roi_align_tdm_kernel<<<grid, 256, 0, stream>>>(feat, rois, out);
}